// _ECELoss_4071628996968
// MI455X (gfx1250) — compile-verified
//
#include <hip/hip_runtime.h>

#define NBINS 15
#define NCLS  128
#define ROWS  40                 // rows per TDM tile: 40*128*4B = 20KB per buffer
#define WAVES 8
#define BLOCK (WAVES * 32)

typedef unsigned int       u32;
typedef unsigned long long u64;
typedef __attribute__((ext_vector_type(4))) u32 v4u;
typedef __attribute__((ext_vector_type(8))) int v8i;
typedef __attribute__((ext_vector_type(4))) int v4i;

// Issue a TDM DMA of a [rows x 128] f32 tile (row-major, stride 128) from
// global memory into LDS at byte offset lds_off. Wave-level op, EXEC ignored,
// tracked by TENSORcnt.
__device__ __forceinline__ void tdm_load_tile(const float* gsrc, u32 lds_off, int rows)
{
    u64 ga = (u64)(uintptr_t)gsrc;
    // D# group 0: count=1 (valid user descriptor), lds_addr, global_addr, type=2
    v4u g0;
    g0.x = 1u;
    g0.y = lds_off;
    g0.z = (u32)ga;                                          // global_addr[31:0]
    g0.w = (u32)((ga >> 32) & 0x01FFFFFFull) | (2u << 30);   // global_addr[56:32] | type=2
    // D# group 1: data_size=4B, tensor_dim0=128, tensor_dim1=rows,
    //             tile_dim0=128, tile_dim1=rows, tensor_dim0_stride=128
    v8i g1;
    g1.s0 = (int)(2u << 16);                                 // data_size=2 (4 bytes)
    g1.s1 = (int)(((u32)NCLS & 0xFFFFu) << 16);              // tensor_dim0[15:0]
    g1.s2 = (int)(((u32)rows & 0xFFFFu) << 16);              // tensor_dim0[31:16]=0 | tensor_dim1[15:0]
    g1.s3 = (int)(((u32)NCLS & 0xFFFFu) << 16);              // tensor_dim1[31:16]=0 | tile_dim0
    g1.s4 = (int)((u32)rows & 0xFFFFu);                      // tile_dim1 | tile_dim2=0
    g1.s5 = NCLS;                                            // tensor_dim0_stride[31:0]
    g1.s6 = 0;                                               // stride0 hi | stride1 lo
    g1.s7 = 0;
    v4i gz4 = {0, 0, 0, 0};                                  // groups 2/3 unused (2-D tensor)
    v8i gz8 = {0, 0, 0, 0, 0, 0, 0, 0};
    __builtin_amdgcn_tensor_load_to_lds(g0, g1, gz4, gz4, gz8, 0);
}

__global__ __launch_bounds__(BLOCK)
void ece_main(const float* __restrict__ logits,
              const int*   __restrict__ labels,
              float*       __restrict__ bins,   // ws: [0..14]=conf_sum, [15..29]=acc_sum
              int n)
{
    __shared__ __align__(16) float buf[2][ROWS * NCLS];
    __shared__ float h_conf[NBINS];
    __shared__ float h_acc[NBINS];

    if (threadIdx.x < NBINS) { h_conf[threadIdx.x] = 0.0f; h_acc[threadIdx.x] = 0.0f; }

    const int wave  = threadIdx.x >> 5;
    const int lane  = threadIdx.x & 31;
    const int tiles = (n + ROWS - 1) / ROWS;

    const u32 lds0 = (u32)(uintptr_t)&buf[0][0];
    const u32 lds1 = (u32)(uintptr_t)&buf[1][0];

    const int t0 = blockIdx.x;
    if (wave == 0 && t0 < tiles) {
        int rows0 = min(ROWS, n - t0 * ROWS);
        tdm_load_tile(logits + (u64)t0 * ROWS * NCLS, lds0, rows0);
    }

    int i = 0;
    for (int t = t0; t < tiles; t += gridDim.x, ++i) {
        const int rowsCur = min(ROWS, n - t * ROWS);
        const int tn = t + (int)gridDim.x;
        if (wave == 0) {
            if (tn < tiles) {
                int rowsN = min(ROWS, n - tn * ROWS);
                // prefetch next tile into the other buffer, then wait for the
                // current tile's DMA (<=1 outstanding => current complete)
                tdm_load_tile(logits + (u64)tn * ROWS * NCLS, (i & 1) ? lds0 : lds1, rowsN);
                __builtin_amdgcn_s_wait_tensorcnt(1);
            } else {
                __builtin_amdgcn_s_wait_tensorcnt(0);
            }
        }
        __syncthreads();   // tile (and histogram init on first pass) visible to all waves

        const float* cur = buf[i & 1];
        for (int r = wave; r < rowsCur; r += WAVES) {
            const float4 v = ((const float4*)(cur + r * NCLS))[lane];

            // joint max / argmax (first-occurrence tie-break like jnp.argmax)
            float m = v.x; int idx = lane * 4;
            if (v.y > m) { m = v.y; idx = lane * 4 + 1; }
            if (v.z > m) { m = v.z; idx = lane * 4 + 2; }
            if (v.w > m) { m = v.w; idx = lane * 4 + 3; }
            #pragma unroll
            for (int off = 16; off > 0; off >>= 1) {
                float om = __shfl_xor(m, off);
                int   oi = __shfl_xor(idx, off);
                if (om > m || (om == m && oi < idx)) { m = om; idx = oi; }
            }

            // sum of exp(x - max); confidence = 1/Z
            float z = __expf(v.x - m) + __expf(v.y - m) + __expf(v.z - m) + __expf(v.w - m);
            #pragma unroll
            for (int off = 16; off > 0; off >>= 1) z += __shfl_xor(z, off);

            if (lane == 0) {
                float conf = 1.0f / z;
                int b = (int)ceilf(conf * (float)NBINS) - 1;
                b = b < 0 ? 0 : (b > NBINS - 1 ? NBINS - 1 : b);
                float acc = (labels[(u64)t * ROWS + r] == idx) ? 1.0f : 0.0f;
                atomicAdd(&h_conf[b], conf);
                atomicAdd(&h_acc[b], acc);
            }
        }
        __syncthreads();   // everyone done reading buf[i&1] before it is overwritten
    }

    if (threadIdx.x < NBINS) {
        atomicAdd(&bins[threadIdx.x],         h_conf[threadIdx.x]);
        atomicAdd(&bins[NBINS + threadIdx.x], h_acc[threadIdx.x]);
    }
}

__global__ void ece_zero(float* bins)
{
    if (threadIdx.x < 2 * NBINS) bins[threadIdx.x] = 0.0f;
}

__global__ void ece_final(const float* __restrict__ bins, float* __restrict__ out, int n)
{
    if (threadIdx.x == 0) {
        float s = 0.0f;
        #pragma unroll
        for (int b = 0; b < NBINS; ++b) s += fabsf(bins[b] - bins[NBINS + b]);
        out[0] = s / (float)n;
    }
}

extern "C" void kernel_launch(void* const* d_in, const int* in_sizes, int n_in,
                              void* d_out, int out_size, void* d_ws, size_t ws_size,
                              hipStream_t stream)
{
    const float* logits = (const float*)d_in[0];
    const int*   labels = (const int*)d_in[1];
    const int n = in_sizes[1];          // number of samples
    float* bins = (float*)d_ws;         // 30 floats of scratch

    ece_zero<<<1, 64, 0, stream>>>(bins);

    int tiles = (n + ROWS - 1) / ROWS;
    int grid  = tiles < 2048 ? tiles : 2048;
    if (grid < 1) grid = 1;
    ece_main<<<grid, BLOCK, 0, stream>>>(logits, labels, bins, n);

    ece_final<<<1, 32, 0, stream>>>(bins, (float*)d_out, n);
}